// PaiConv_2723009266472
// MI455X (gfx1250) — compile-verified
//
#include <hip/hip_runtime.h>

typedef __attribute__((ext_vector_type(2))) float v2f;
typedef __attribute__((ext_vector_type(8))) float v8f;

constexpr int BB      = 8;      // batch
constexpr int NPTS    = 20000;  // points
constexpr int KNB     = 9;      // neighbors
constexpr int CH      = 64;     // in_c == out_c
constexpr int KC      = KNB * CH;  // 576 = conv K-dim
constexpr int TPTS    = 16;     // points per tile (WMMA M)
constexpr int THREADS = 128;    // 4 waves

// LDS row strides padded to kill bank conflicts on ds_load_b64 of A fragments:
// stride mod 64 == 2 -> 16 lanes (rows) hit distinct bank pairs.
constexpr int MSTRIDE = 578;    // mixed tile [16][576] padded
constexpr int XSTRIDE = 66;     // x tile [16][64] padded

__launch_bounds__(THREADS)
__global__ void paiconv_kernel(const float* __restrict__ x,
                               const long long* __restrict__ nidx,
                               const float* __restrict__ adjw,
                               const float* __restrict__ conv_w,
                               const float* __restrict__ conv_b,
                               const float* __restrict__ mlp_w,
                               const float* __restrict__ mlp_b,
                               float* __restrict__ out)
{
    __shared__ __align__(16) float mixedl[TPTS * MSTRIDE]; // 16x576 mixed+elu tile (A for conv GEMM)
    __shared__ __align__(16) float adjl[TPTS * 81];        // per-point 9x9 adjweight
    __shared__ __align__(16) float xt[TPTS * XSTRIDE];     // masked 16x64 x tile (A for mlp GEMM)
    __shared__ int nbrl[TPTS * KNB];                       // neighbor indices

    const int tid  = threadIdx.x;
    const int tile = blockIdx.x;       // 0..1249
    const int b    = blockIdx.y;       // 0..7
    const size_t xbase = (size_t)b * NPTS * CH;

    // ---- stage adjweight tile + neighbor indices ----
    for (int j = tid; j < TPTS * 81; j += THREADS) {
        int p = j / 81, r = j - p * 81;
        adjl[j] = adjw[(size_t)(tile * TPTS + p) * 81 + r];
    }
    for (int j = tid; j < TPTS * KNB; j += THREADS) {
        int p = j / KNB, k = j - p * KNB;
        nbrl[j] = (int)nidx[((size_t)b * NPTS + tile * TPTS + p) * KNB + k];
    }
    __syncthreads();

    // ---- gather (L2-served) + 9x9 adj mix + ELU -> mixedl; stage masked x tile ----
    #pragma unroll
    for (int i = 0; i < (TPTS * CH) / THREADS; ++i) {      // 8 iterations
        int pair = tid + i * THREADS;                      // (p, f) pair
        int p = pair >> 6;
        int f = pair & 63;
        int n = tile * TPTS + p;
        // residual-path input (x * mask)
        xt[p * XSTRIDE + f] = (n == NPTS - 1) ? 0.f : x[xbase + (size_t)n * CH + f];
        float xk[KNB];
        #pragma unroll
        for (int k = 0; k < KNB; ++k) {
            int nb = nbrl[p * KNB + k];
            xk[k] = (nb == NPTS - 1) ? 0.f : x[xbase + (size_t)nb * CH + f];
        }
        #pragma unroll
        for (int t = 0; t < KNB; ++t) {
            float acc = 0.f;
            #pragma unroll
            for (int k = 0; k < KNB; ++k)
                acc = fmaf(xk[k], adjl[p * 81 + k * KNB + t], acc);
            acc = acc > 0.f ? acc : expm1f(acc);           // ELU
            mixedl[p * MSTRIDE + t * CH + f] = acc;        // reshape order: t*64+f
        }
    }
    __syncthreads();

    // ---- WMMA phase: each wave owns a 16-wide slice of the 64 output channels ----
    const int wave = tid >> 5;
    const int lane = tid & 31;
    const int hi   = lane >> 4;        // K half-select for A/B fragments
    const int lr   = lane & 15;        // A row (M) / B column (N)
    const int o    = wave * 16 + lr;   // output channel for this lane's B column

    // conv GEMM: [16 x 576] @ B[576 x 16], B[k][o] = conv_w[o*576 + k]
    const float* bw = conv_w + (size_t)o * KC;
    v8f acc0 = {}; v8f acc1 = {};
    #pragma unroll 4
    for (int kk = 0; kk < KC / 4; kk += 2) {               // 144 WMMAs, 2 acc chains
        int k0 = kk * 4 + 2 * hi;
        v2f a0 = *(const v2f*)&mixedl[lr * MSTRIDE + k0];
        v2f b0 = *(const v2f*)&bw[k0];
        acc0 = __builtin_amdgcn_wmma_f32_16x16x4_f32(false, a0, false, b0,
                                                     (short)0, acc0, false, false);
        int k1 = k0 + 4;
        v2f a1 = *(const v2f*)&mixedl[lr * MSTRIDE + k1];
        v2f b1 = *(const v2f*)&bw[k1];
        acc1 = __builtin_amdgcn_wmma_f32_16x16x4_f32(false, a1, false, b1,
                                                     (short)0, acc1, false, false);
    }
    v8f convacc = acc0 + acc1;

    // residual MLP GEMM: [16 x 64] @ B[64 x 16], B[k][o] = mlp_w[o*64 + k]
    const float* mw = mlp_w + (size_t)o * CH;
    v8f acc2 = {};
    #pragma unroll 4
    for (int kk = 0; kk < CH / 4; ++kk) {                  // 16 WMMAs
        int k0 = kk * 4 + 2 * hi;
        v2f a = *(const v2f*)&xt[lr * XSTRIDE + k0];
        v2f bv = *(const v2f*)&mw[k0];
        acc2 = __builtin_amdgcn_wmma_f32_16x16x4_f32(false, a, false, bv,
                                                     (short)0, acc2, false, false);
    }

    // ---- epilogue: bias + ELU + mask on conv path, add residual path, store ----
    const float cb = conv_b[o];
    const float mb = mlp_b[o];
    float* outp = out + ((size_t)b * NPTS + tile * TPTS) * CH;
    #pragma unroll
    for (int r = 0; r < 8; ++r) {
        int row = r + 8 * hi;                              // C/D layout: M = r + 8*hi
        int n   = tile * TPTS + row;
        float v = convacc[r] + cb;
        v = v > 0.f ? v : expm1f(v);
        if (n == NPTS - 1) v = 0.f;                        // out_feat * mask
        outp[(size_t)row * CH + o] = v + acc2[r] + mb;
    }
}

extern "C" void kernel_launch(void* const* d_in, const int* in_sizes, int n_in,
                              void* d_out, int out_size, void* d_ws, size_t ws_size,
                              hipStream_t stream)
{
    const float*     x      = (const float*)d_in[0];
    const long long* ni     = (const long long*)d_in[1];   // int64 neighbor_index
    const float*     adjw   = (const float*)d_in[2];
    const float*     conv_w = (const float*)d_in[3];
    const float*     conv_b = (const float*)d_in[4];
    const float*     mlp_w  = (const float*)d_in[5];
    const float*     mlp_b  = (const float*)d_in[6];
    float* out = (float*)d_out;

    dim3 grid(NPTS / TPTS, BB);   // 1250 x 8 workgroups
    paiconv_kernel<<<grid, THREADS, 0, stream>>>(x, ni, adjw, conv_w, conv_b,
                                                 mlp_w, mlp_b, out);
    (void)in_sizes; (void)n_in; (void)out_size; (void)d_ws; (void)ws_size;
}